// AdEx_52201032516273
// MI455X (gfx1250) — compile-verified
//
#include <hip/hip_runtime.h>

// AdEx RHS, memory-bound streaming kernel for MI455X (gfx1250, wave32).
//   dVdt = (-(V - V_rest) + delta_T*exp((V - V_T)/delta_T) - R*w + R*I(t)) / tau
//   dwdt = (a*(V - V_rest) - w) / tau_w
// Folded form (constants computed once per thread from uniform scalar loads):
//   e     = exp2( fma(V, kE, bE) )                         kE =  log2e/delta_T
//                                                          bE = -V_T*log2e/delta_T
//   dVdt  = fma(-R/tau, w, fma(delta_T/tau, e, fma(-1/tau, V, (V_rest + R*I_t)/tau)))
//   dwdt  = fma(a/tau_w, V, fma(-1/tau_w, w, -a*V_rest/tau_w))

typedef __attribute__((ext_vector_type(4))) float vfloat4;

__device__ __forceinline__ void adex_consts(
    const float* __restrict__ t, const float* __restrict__ I_ext,
    const float* __restrict__ V_rest, const float* __restrict__ V_T,
    const float* __restrict__ delta_T, const float* __restrict__ R,
    const float* __restrict__ tau, const float* __restrict__ tau_w,
    const float* __restrict__ a,
    float& c0, float& n_itau, float& cE, float& kE, float& bE, float& n_cW,
    float& cA, float& c1, float& n_itw)
{
    // All uniform-address loads: cheap, cached in WGP$/L2 (RT temporal hint).
    float t0 = t[0];
    int idx = (int)floorf(t0 / 0.001f);          // match reference t // 0.001
    idx = idx < 0 ? 0 : (idx > 999 ? 999 : idx);
    float I_t = I_ext[idx];

    float Vr   = V_rest[0];
    float VT   = V_T[0];
    float dT   = delta_T[0];
    float Rv   = R[0];
    float itau = 1.0f / tau[0];
    float itw  = 1.0f / tau_w[0];
    float av   = a[0];

    const float LOG2E = 1.44269504088896340736f;
    kE     = LOG2E / dT;
    bE     = -VT * kE;
    cE     = dT * itau;
    c0     = (Vr + Rv * I_t) * itau;
    n_cW   = -(Rv * itau);
    n_itau = -itau;
    cA     = av * itw;
    c1     = -av * Vr * itw;
    n_itw  = -itw;
}

__global__ __launch_bounds__(256) void adex_rhs_v4(
    const float* __restrict__ V, const float* __restrict__ w,
    const float* __restrict__ t, const float* __restrict__ I_ext,
    const float* __restrict__ V_rest, const float* __restrict__ V_T,
    const float* __restrict__ delta_T, const float* __restrict__ R,
    const float* __restrict__ tau, const float* __restrict__ tau_w,
    const float* __restrict__ a,
    float* __restrict__ dV, float* __restrict__ dw, int n4)
{
    int i = blockIdx.x * blockDim.x + threadIdx.x;
    if (i >= n4) return;

    float c0, n_itau, cE, kE, bE, n_cW, cA, c1, n_itw;
    adex_consts(t, I_ext, V_rest, V_T, delta_T, R, tau, tau_w, a,
                c0, n_itau, cE, kE, bE, n_cW, cA, c1, n_itw);

    // Streaming 128-bit non-temporal loads: data is read exactly once
    // (512 MB total traffic > 192 MB L2), so NT avoids cache pollution.
    const vfloat4* V4 = (const vfloat4*)V;
    const vfloat4* W4 = (const vfloat4*)w;
    vfloat4 v  = __builtin_nontemporal_load(V4 + i);
    vfloat4 ww = __builtin_nontemporal_load(W4 + i);

    vfloat4 dv, dwv;
#pragma unroll
    for (int k = 0; k < 4; ++k) {
        float vk = v[k];
        float wk = ww[k];
        float e  = exp2f(fmaf(vk, kE, bE));                     // v_exp_f32 (TRANS pipe)
        dv[k]  = fmaf(n_cW, wk, fmaf(cE, e, fmaf(n_itau, vk, c0)));
        dwv[k] = fmaf(cA, vk, fmaf(n_itw, wk, c1));
    }

    // Write-once outputs: non-temporal 128-bit streaming stores.
    __builtin_nontemporal_store(dv,  (vfloat4*)dV + i);
    __builtin_nontemporal_store(dwv, (vfloat4*)dw + i);
}

// Scalar tail (not used for N = 2^25, kept for generality).
__global__ __launch_bounds__(64) void adex_rhs_tail(
    const float* __restrict__ V, const float* __restrict__ w,
    const float* __restrict__ t, const float* __restrict__ I_ext,
    const float* __restrict__ V_rest, const float* __restrict__ V_T,
    const float* __restrict__ delta_T, const float* __restrict__ R,
    const float* __restrict__ tau, const float* __restrict__ tau_w,
    const float* __restrict__ a,
    float* __restrict__ dV, float* __restrict__ dw, int start, int n)
{
    int i = start + blockIdx.x * blockDim.x + threadIdx.x;
    if (i >= n) return;

    float c0, n_itau, cE, kE, bE, n_cW, cA, c1, n_itw;
    adex_consts(t, I_ext, V_rest, V_T, delta_T, R, tau, tau_w, a,
                c0, n_itau, cE, kE, bE, n_cW, cA, c1, n_itw);

    float vk = V[i];
    float wk = w[i];
    float e  = exp2f(fmaf(vk, kE, bE));
    dV[i] = fmaf(n_cW, wk, fmaf(cE, e, fmaf(n_itau, vk, c0)));
    dw[i] = fmaf(cA, vk, fmaf(n_itw, wk, c1));
}

extern "C" void kernel_launch(void* const* d_in, const int* in_sizes, int n_in,
                              void* d_out, int out_size, void* d_ws, size_t ws_size,
                              hipStream_t stream) {
    // setup_inputs() order:
    // 0:t 1:V 2:w 3:I_ext 4:V_rest 5:V_T 6:delta_T 7:R 8:tau 9:tau_w 10:a
    const float* t       = (const float*)d_in[0];
    const float* V       = (const float*)d_in[1];
    const float* w       = (const float*)d_in[2];
    const float* I_ext   = (const float*)d_in[3];
    const float* V_rest  = (const float*)d_in[4];
    const float* V_T     = (const float*)d_in[5];
    const float* delta_T = (const float*)d_in[6];
    const float* R       = (const float*)d_in[7];
    const float* tau     = (const float*)d_in[8];
    const float* tau_w   = (const float*)d_in[9];
    const float* a       = (const float*)d_in[10];

    const int n = in_sizes[1];           // N = 33554432
    float* dV = (float*)d_out;           // tuple return order: dVdt first
    float* dw = dV + n;

    const int n4 = n >> 2;
    if (n4 > 0) {
        const int block = 256;           // 8 wave32 waves per block
        const int grid  = (n4 + block - 1) / block;
        adex_rhs_v4<<<grid, block, 0, stream>>>(
            V, w, t, I_ext, V_rest, V_T, delta_T, R, tau, tau_w, a, dV, dw, n4);
    }
    const int rem = n & 3;
    if (rem > 0) {
        adex_rhs_tail<<<1, 64, 0, stream>>>(
            V, w, t, I_ext, V_rest, V_T, delta_T, R, tau, tau_w, a, dV, dw, n4 << 2, n);
    }
}